// QonvLayer_last_67499706024572
// MI455X (gfx1250) — compile-verified
//
#include <hip/hip_runtime.h>
#include <cstdint>
#include <cstddef>

// ---------------------------------------------------------------------------
// QonvLayer for MI455X (gfx1250, wave32).
//   out[b, ci*16+c, j, k, l] = psi(cube)^T * A_ci * psi(cube),  j,k,l in [0,16)
// where A_ci = Re(U_ci^dag Z0 U_ci) is a 16x16 real matrix built once from the
// weights, and psi is the rank-1 real product state from the 8 cube corners.
// Memory-bound (~26 MiB total traffic -> ~1.1 us @ 23.3 TB/s); the per-tile
// 16x16x16 GEMM is done with V_WMMA_F32_16X16X4_F32 (f32 => reference precision).
// ---------------------------------------------------------------------------

typedef __attribute__((ext_vector_type(2))) float v2f;
typedef __attribute__((ext_vector_type(8))) float v8f;

#ifndef __has_builtin
#define __has_builtin(x) 0
#endif
#if __has_builtin(__builtin_amdgcn_wmma_f32_16x16x4_f32)
#define USE_WMMA_F32 1
#else
#define USE_WMMA_F32 0
#endif

#define PI_F 3.14159265358979323846f

// Packed gate plan (<=64 ops).  bit0: 1=CNOT, 0=rotation.
//   rot : gate=(op>>1)&3 (0=RX,1=RY,2=RZ), wire=(op>>3)&3, widx=(op>>5)&7
//   cnot: ctrl=(op>>1)&3, tgt=(op>>3)&3
struct GatePlan { int n; unsigned ops[64]; };

// ============================ host: numpy RNG ==============================
// Best-effort replication of np.random.default_rng(1234):
// SeedSequence -> PCG64 (XSL-RR), Generator.random(), Lemire bounded integers
// (Generator.integers), and masked-rejection random_interval with the 32-bit
// draw buffer (Generator.shuffle, used by choice(replace=False)).
namespace nprng {
typedef unsigned __int128 u128;
struct Pcg64 { u128 state, inc; bool has32; uint32_t buf32; };
static inline u128 pcg_mult() {
  return (((u128)0x2360ed051fc65da4ULL) << 64) | 0x4385df649fccf645ULL;
}
static inline uint64_t rotr64(uint64_t v, unsigned r) {
  return (v >> r) | (v << ((64u - r) & 63u));
}
static inline uint64_t next64(Pcg64& g) {
  g.state = g.state * pcg_mult() + g.inc;
  uint64_t hi = (uint64_t)(g.state >> 64), lo = (uint64_t)g.state;
  return rotr64(hi ^ lo, (unsigned)(hi >> 58));
}
static inline uint32_t next32(Pcg64& g) {           // numpy's buffered next32
  if (g.has32) { g.has32 = false; return g.buf32; }
  uint64_t n = next64(g);
  g.has32 = true; g.buf32 = (uint32_t)(n >> 32);
  return (uint32_t)n;
}
static inline double nextdouble(Pcg64& g) {
  return (double)(next64(g) >> 11) * (1.0 / 9007199254740992.0);
}
static void seed_pcg(Pcg64& g, uint32_t entropy) {
  uint32_t pool[4]; uint32_t hc = 0x43b0d7e5u;
  auto hashmix = [&hc](uint32_t v) {
    v ^= hc; hc *= 0x931e8875u; v *= hc; v ^= v >> 16; return v;
  };
  auto mix = [](uint32_t x, uint32_t y) {
    uint32_t r = 0xca01f9ddu * x - 0x4973f715u * y; r ^= r >> 16; return r;
  };
  for (int i = 0; i < 4; ++i) pool[i] = hashmix(i == 0 ? entropy : 0u);
  for (int s = 0; s < 4; ++s)
    for (int d = 0; d < 4; ++d)
      if (s != d) pool[d] = mix(pool[d], hashmix(pool[s]));
  uint32_t hb = 0x8b51f9ddu; uint32_t w[8];
  for (int i = 0; i < 8; ++i) {
    uint32_t v = pool[i & 3];
    v ^= hb; hb *= 0x58f38dedu; v *= hb; v ^= v >> 16; w[i] = v;
  }
  uint64_t s64[4];
  for (int i = 0; i < 4; ++i) s64[i] = (uint64_t)w[2 * i] | ((uint64_t)w[2 * i + 1] << 32);
  u128 initstate = (((u128)s64[0]) << 64) | s64[1];
  u128 initseq   = (((u128)s64[2]) << 64) | s64[3];
  g.state = 0; g.inc = (initseq << 1) | 1; g.has32 = false; g.buf32 = 0;
  g.state = g.state * pcg_mult() + g.inc;
  g.state += initstate;
  g.state = g.state * pcg_mult() + g.inc;
}
static uint64_t lemire(Pcg64& g, uint64_t rng) {     // integers(0, rng+1)
  uint64_t rng_excl = rng + 1;
  u128 m = (u128)next64(g) * rng_excl;
  uint64_t leftover = (uint64_t)m;
  if (leftover < rng_excl) {
    uint64_t threshold = (uint64_t)(0 - rng_excl) % rng_excl;
    while (leftover < threshold) { m = (u128)next64(g) * rng_excl; leftover = (uint64_t)m; }
  }
  return (uint64_t)(m >> 64);
}
static uint64_t interval(Pcg64& g, uint64_t mx) {    // shuffle's random_interval
  if (mx == 0) return 0;
  uint64_t mask = mx;
  mask |= mask >> 1; mask |= mask >> 2; mask |= mask >> 4;
  mask |= mask >> 8; mask |= mask >> 16; mask |= mask >> 32;
  uint64_t v;
  while ((v = (uint64_t)(next32(g) & (uint32_t)mask)) > mx) {}
  return v;
}
}  // namespace nprng

static GatePlan build_plan() {
  GatePlan p; p.n = 0;
  nprng::Pcg64 g; nprng::seed_pcg(g, 1234u);
  for (int l = 0; l < 2; ++l) {
    int i = 0;
    while (i < 4) {
      if (p.n >= 64) return p;                    // safety cap
      if (nprng::nextdouble(g) > 0.3) {           // rotation
        int gate = (int)nprng::lemire(g, 2);      // integers(3)
        int wire = (int)nprng::lemire(g, 3);      // integers(4)
        int widx = l * 4 + i;
        p.ops[p.n++] = (unsigned)((gate << 1) | (wire << 3) | (widx << 5));
        ++i;
      } else {                                    // choice(4,2,replace=False)
        int arr[4] = {0, 1, 2, 3};                //   = permutation(4)[:2]
        for (int ii = 3; ii >= 1; --ii) {
          int jj = (int)nprng::interval(g, (uint64_t)ii);
          int t = arr[ii]; arr[ii] = arr[jj]; arr[jj] = t;
        }
        p.ops[p.n++] = 1u | ((unsigned)arr[0] << 1) | ((unsigned)arr[1] << 3);
      }
    }
  }
  return p;
}

// ====================== kernel 0: zero-fill output =========================
__global__ void qonv_zero(float4* __restrict__ o, int n4) {
  const float4 z = make_float4(0.f, 0.f, 0.f, 0.f);
  for (int i = blockIdx.x * blockDim.x + threadIdx.x; i < n4;
       i += gridDim.x * blockDim.x)
    o[i] = z;
}

// ============ kernel 1: build A_ci = Re(U^dag Z0 U), 16x16 each ============
__global__ __launch_bounds__(32) void qonv_build_A(
    const float* __restrict__ w1, const float* __restrict__ w2,
    float* __restrict__ Aout, GatePlan plan) {
  __shared__ float Ur[2][16][16], Ui[2][16][16];
  const int tid = threadIdx.x;            // 32 threads: 2 circuits x 16 columns
  const int which = tid >> 4, col = tid & 15;
  const float* wt = which ? w2 : w1;      // weights[l,i] flat, widx = l*4+i
  float sr[16], si[16];
#pragma unroll
  for (int k = 0; k < 16; ++k) { sr[k] = (k == col) ? 1.f : 0.f; si[k] = 0.f; }

  for (int o = 0; o < plan.n; ++o) {
    const unsigned op = plan.ops[o];
    if (op & 1u) {                                         // CNOT
      const int cm = 8 >> ((op >> 1) & 3), tm = 8 >> ((op >> 3) & 3);
      for (int k = 0; k < 16; ++k)
        if ((k & cm) && !(k & tm)) {
          const int k2 = k | tm;
          float tr = sr[k], ti = si[k];
          sr[k] = sr[k2]; si[k] = si[k2]; sr[k2] = tr; si[k2] = ti;
        }
    } else {                                               // RX/RY/RZ
      const int gate = (op >> 1) & 3, m = 8 >> ((op >> 3) & 3);
      const float th = wt[(op >> 5) & 7] * 0.5f;
      float s_, c_; sincosf(th, &s_, &c_);
      for (int k = 0; k < 16; ++k)
        if (!(k & m)) {
          const int k2 = k | m;
          const float a0r = sr[k], a0i = si[k], a1r = sr[k2], a1i = si[k2];
          if (gate == 0) {        // RX: [[c,-is],[-is,c]]
            sr[k]  =  c_ * a0r + s_ * a1i;  si[k]  =  c_ * a0i - s_ * a1r;
            sr[k2] =  s_ * a0i + c_ * a1r;  si[k2] = -s_ * a0r + c_ * a1i;
          } else if (gate == 1) { // RY: [[c,-s],[s,c]]
            sr[k]  =  c_ * a0r - s_ * a1r;  si[k]  =  c_ * a0i - s_ * a1i;
            sr[k2] =  s_ * a0r + c_ * a1r;  si[k2] =  s_ * a0i + c_ * a1i;
          } else {                // RZ: diag(e^{-i t/2}, e^{+i t/2})
            sr[k]  =  c_ * a0r + s_ * a0i;  si[k]  =  c_ * a0i - s_ * a0r;
            sr[k2] =  c_ * a1r - s_ * a1i;  si[k2] =  c_ * a1i + s_ * a1r;
          }
        }
    }
  }
#pragma unroll
  for (int j = 0; j < 16; ++j) { Ur[which][j][col] = sr[j]; Ui[which][j][col] = si[j]; }
  __syncthreads();
  // A[k][kp] = sum_j z_j * Re(conj(U[j,k]) U[j,kp]),  z_j = +1 (j<8) / -1
  for (int kp = 0; kp < 16; ++kp) {
    float acc = 0.f;
#pragma unroll
    for (int j = 0; j < 16; ++j) {
      const float z = (j < 8) ? 1.f : -1.f;
      acc += z * (Ur[which][j][col] * Ur[which][j][kp] +
                  Ui[which][j][col] * Ui[which][j][kp]);
    }
    Aout[which * 256 + col * 16 + kp] = acc;
  }
}

// ======================= kernel 2: main WMMA kernel ========================
// One wave32 per output row of 16 cubes: wid -> (b, c, j, k), l = 0..15.
__global__ __launch_bounds__(256) void qonv_main(
    const float* __restrict__ x, const float* __restrict__ Amat,
    float* __restrict__ out) {
  __shared__ float sA[2][16][16];          // the two circuit matrices
  __shared__ float cs[8][16][8][2];        // [wave][cube][corner s8][cos/sin]
  __shared__ float psi[8][2][16][17];      // padded x17: conflict-free on 64 banks
  __shared__ float prod[8][16][17];

  const int tid = threadIdx.x;
  const int w = tid >> 5, lane = tid & 31;         // wave32 (gfx1250)
  for (int e = tid; e < 512; e += 256) ((float*)sA)[e] = Amat[e];

  const int wid = blockIdx.x * 8 + w;
  const int kk = wid & 15, jj = (wid >> 4) & 15;
  const int c = (wid >> 8) & 15, b = wid >> 12;
  const float* xb = x + ((size_t)(b * 16 + c) << 18);   // 64^3 per (b,c)

  // --- load the 4 (dj,dk) input rows (32 floats each): one b128 per lane ---
  {
    const int r = lane >> 3, dj = r >> 1, dk = r & 1, w4 = (lane & 7) << 2;
    const size_t off = ((size_t)(2 * jj + dj) * 64 + (size_t)(2 * kk + dk)) * 64 + w4;
    const float4 v = *(const float4*)(xb + off);
    const float ys[4] = {v.x, v.y, v.z, v.w};
#pragma unroll
    for (int e = 0; e < 4; ++e) {
      const int wpos = w4 + e, cube = wpos >> 1, dl = wpos & 1;
      const int s8 = dj * 4 + dk * 2 + dl;
      const float a = 0.5f * fminf(PI_F, PI_F * ys[e]);  // min(pi, pi*y)/2
      float sn, cn; __sincosf(a, &sn, &cn);
      cs[w][cube][s8][0] = cn; cs[w][cube][s8][1] = sn;
    }
  }
  __syncthreads();

  // --- build the two 16x16 product-state matrices (rank-1 Kronecker rows) ---
  // state index q: wire0->bit3 ... wire3->bit0; circuit1 corners {0,6,3,5},
  // circuit2 corners {7,1,4,2} (parity classes of (dj,dk,dl)).
#pragma unroll
  for (int t = 0; t < 8; ++t) {
    const int e = lane * 8 + t, m = e >> 4, q = e & 15;
    const int b3 = (q >> 3) & 1, b2 = (q >> 2) & 1, b1 = (q >> 1) & 1, b0 = q & 1;
    psi[w][0][m][q] = cs[w][m][0][b3] * cs[w][m][6][b2] *
                      cs[w][m][3][b1] * cs[w][m][5][b0];
    psi[w][1][m][q] = cs[w][m][7][b3] * cs[w][m][1][b2] *
                      cs[w][m][4][b1] * cs[w][m][2][b0];
  }
  __syncthreads();

  const int half = lane >> 4, col = lane & 15;
#pragma unroll
  for (int ci = 0; ci < 2; ++ci) {
#if USE_WMMA_F32
    // P = Psi * A via 4 accumulating V_WMMA_F32_16X16X4_F32.
    // A-frag (16x4 f32): v0 lanes0-15 K=0 / lanes16-31 K=2; v1: K=1 / K=3.
    v8f acc = {0.f, 0.f, 0.f, 0.f, 0.f, 0.f, 0.f, 0.f};
#pragma unroll
    for (int t = 0; t < 4; ++t) {
      v2f af, bf;
      af.x = psi[w][ci][col][4 * t + 2 * half + 0];
      af.y = psi[w][ci][col][4 * t + 2 * half + 1];
      bf.x = sA[ci][4 * t + 2 * half + 0][col];
      bf.y = sA[ci][4 * t + 2 * half + 1][col];
      acc = __builtin_amdgcn_wmma_f32_16x16x4_f32(
          /*neg_a=*/false, af, /*neg_b=*/false, bf,
          /*c_mod=*/(short)0, acc, /*reuse_a=*/false, /*reuse_b=*/false);
    }
    // C/D layout: VGPR r, lane -> (M = r + 8*half, N = col)
#pragma unroll
    for (int rr = 0; rr < 8; ++rr) {
      const int m = rr + 8 * half;
      prod[w][m][col] = acc[rr] * psi[w][ci][m][col];
    }
#else
    // VALU fallback (same math), used only if the f32 WMMA builtin is absent.
#pragma unroll
    for (int rr = 0; rr < 8; ++rr) {
      const int m = rr + 8 * half;
      float s = 0.f;
#pragma unroll
      for (int k = 0; k < 16; ++k) s += psi[w][ci][m][k] * sA[ci][k][col];
      prod[w][m][col] = s * psi[w][ci][m][col];
    }
#endif
    __syncthreads();
    if (lane < 16) {                       // row-reduce + coalesced 16-lane store
      float ssum = 0.f;
#pragma unroll
      for (int n = 0; n < 16; ++n) ssum += prod[w][lane][n];
      const size_t oidx =
          (((size_t)(b * 32 + ci * 16 + c) * 32 + jj) * 32 + kk) * 32 + lane;
      out[oidx] = ssum;
    }
    __syncthreads();                       // prod reused by second circuit
  }
}

// ================================ launch ===================================
extern "C" void kernel_launch(void* const* d_in, const int* in_sizes, int n_in,
                              void* d_out, int out_size, void* d_ws, size_t ws_size,
                              hipStream_t stream) {
  (void)in_sizes; (void)n_in; (void)ws_size;
  const float* x  = (const float*)d_in[0];   // (4,16,64,64,64) f32
  const float* w1 = (const float*)d_in[1];   // (2,4) f32
  const float* w2 = (const float*)d_in[2];   // (2,4) f32
  float* out  = (float*)d_out;               // (4,32,32,32,32) f32
  float* Amat = (float*)d_ws;                // 2 x 16 x 16 f32 scratch

  const GatePlan plan = build_plan();        // deterministic, host-side

  qonv_zero<<<2048, 256, 0, stream>>>((float4*)out, out_size / 4);
  qonv_build_A<<<1, 32, 0, stream>>>(w1, w2, Amat, plan);
  qonv_main<<<2048, 256, 0, stream>>>(x, Amat, out);
}